// AttnDecoder_24627342475413
// MI455X (gfx1250) — compile-verified
//
#include <hip/hip_runtime.h>
#include <math.h>

// ---------------------------------------------------------------------------
// Problem constants (match reference)
// ---------------------------------------------------------------------------
enum : int {
  VOCAB_ = 32000, EMB_ = 256, ENC_ = 512, DEC_ = 512, ATTN_ = 256,
  B_ = 32, T_ = 128, U_ = 64,
  XDIM_ = DEC_ + ENC_   // 1024
};

typedef float v2f __attribute__((ext_vector_type(2)));
typedef float v8f __attribute__((ext_vector_type(8)));

// ---------------------------------------------------------------------------
// CDNA5 async global->LDS copy (ASYNCcnt-tracked DMA, no VGPR round trip).
// LDS destination address = low 32 bits of the generic shared-memory pointer.
// ---------------------------------------------------------------------------
__device__ __forceinline__ void async_copy_b128(void* lds_dst, const void* gsrc) {
  unsigned loff = (unsigned)(uintptr_t)lds_dst;
  asm volatile("global_load_async_to_lds_b128 %0, %1, off"
               :: "v"(loff), "v"(gsrc)
               : "memory");
}
__device__ __forceinline__ void wait_asynccnt0() {
  asm volatile("s_wait_asynccnt 0" ::: "memory");
}

// ---------------------------------------------------------------------------
// WMMA fp32 GEMM:  C[M,N] = A[M,K] @ B  (+ bias)
//   BT == false : B is row-major K x N   (LDS tile layout [k][n])
//   BT == true  : B is row-major N x K   (A @ B^T; LDS tile layout [n][k]
//                 so both global reads and LDS writes are contiguous B128)
// 256 threads (8 waves). Block tile 64(M) x 256(N) x 32(K).
// Waves 2(M) x 4(N); each wave owns a 32x64 strip = 8 accumulators, so each
// B fragment is reused by two WMMAs (8 wmma : 6 ds-loads per k-step).
// Double-buffered LDS; next K-chunk staged with async DMA while WMMAs run.
// ---------------------------------------------------------------------------
#define BM 64
#define BN 256
#define BK 32

template <bool BT>
__global__ __launch_bounds__(256)
void wmma_gemm_f32(const float* __restrict__ A, const float* __restrict__ Bm,
                   const float* __restrict__ bias, float* __restrict__ C,
                   int M, int N, int K)
{
  constexpr int BROWS = BT ? BN : BK;
  constexpr int BCOLS = BT ? (BK + 4) : (BN + 4);

  __shared__ float As[2][BM][BK + 4];
  __shared__ float Bs[2][BROWS][BCOLS];

  const int tid  = threadIdx.x;
  const int lane = tid & 31;
  const int wid  = tid >> 5;     // 0..7
  const int wy   = wid >> 2;     // 0..1  (M)
  const int wx   = wid & 3;      // 0..3  (N)

  const int m0 = blockIdx.y * BM;
  const int n0 = blockIdx.x * BN;

  const int lrow = lane & 15;    // 0..15
  const int lhi  = lane >> 4;    // 0 or 1 (lane-half)

  v8f acc[2][4] = {};            // [m-subtile][n-subtile] 16x16 f32 accums

  // ---- async stage of one K-chunk into LDS buffer `bf` -------------------
  auto stage = [&](int bf, int k0) {
    {   // A tile: 64x32, two contiguous float4 copies per thread
      const int r  = tid >> 3;            // 0..31
      const int cc = (tid & 7) << 2;      // 0,4,...,28
      async_copy_b128(&As[bf][r][cc],
                      A + (size_t)(m0 + r) * K + k0 + cc);
      async_copy_b128(&As[bf][r + 32][cc],
                      A + (size_t)(m0 + r + 32) * K + k0 + cc);
    }
    if constexpr (BT) {
      // B tile stored [n][k]: thread n copies its 32 consecutive k-values
      const int n = tid;                  // 0..255
      const float* src = Bm + (size_t)(n0 + n) * K + k0;
#pragma unroll
      for (int i = 0; i < 8; ++i)
        async_copy_b128(&Bs[bf][n][i * 4], src + i * 4);
    } else {
      // B tile stored [k][n]: contiguous rows of the K x N matrix
      const int cc = (tid & 63) << 2;     // col 0..252
      const int rb = tid >> 6;            // 0..3
#pragma unroll
      for (int i = 0; i < 8; ++i) {
        const int r = rb + i * 4;
        async_copy_b128(&Bs[bf][r][cc], Bm + (size_t)(k0 + r) * N + n0 + cc);
      }
    }
  };

  int buf = 0;
  stage(0, 0);
  wait_asynccnt0();
  __syncthreads();

  for (int k0 = 0; k0 < K; k0 += BK) {
    if (k0 + BK < K) stage(buf ^ 1, k0 + BK);   // overlap DMA with compute

    const int arow = wy * 32 + lrow;
#pragma unroll
    for (int kk = 0; kk < BK / 4; ++kk) {
      const int ka = kk * 4 + lhi * 2;    // this lane-half's K pair
      v2f a0, a1;
      a0.x = As[buf][arow][ka + 0];
      a0.y = As[buf][arow][ka + 1];
      a1.x = As[buf][arow + 16][ka + 0];
      a1.y = As[buf][arow + 16][ka + 1];
#pragma unroll
      for (int t = 0; t < 4; ++t) {
        const int col = wx * 64 + t * 16 + lrow;
        v2f bfr;
        if constexpr (BT) {
          bfr.x = Bs[buf][col][ka + 0];   // contiguous -> ds_load_b64
          bfr.y = Bs[buf][col][ka + 1];
        } else {
          bfr.x = Bs[buf][ka + 0][col];
          bfr.y = Bs[buf][ka + 1][col];
        }
        acc[0][t] = __builtin_amdgcn_wmma_f32_16x16x4_f32(
            false, a0, false, bfr, (short)0, acc[0][t], false, false);
        acc[1][t] = __builtin_amdgcn_wmma_f32_16x16x4_f32(
            false, a1, false, bfr, (short)0, acc[1][t], false, false);
      }
    }

    wait_asynccnt0();   // my async copies for the next chunk are done
    __syncthreads();    // everyone finished compute + staging
    buf ^= 1;
  }

  // ---- epilogue: C/D layout — VGPR r: lanes0-15 M=r, lanes16-31 M=r+8 ----
#pragma unroll
  for (int mt = 0; mt < 2; ++mt) {
#pragma unroll
    for (int t = 0; t < 4; ++t) {
      const int col = n0 + wx * 64 + t * 16 + lrow;
      const float bv = bias ? bias[col] : 0.f;
#pragma unroll
      for (int r = 0; r < 8; ++r) {
        const int row = m0 + wy * 32 + mt * 16 + r + lhi * 8;
        C[(size_t)row * N + col] = acc[mt][t][r] + bv;
      }
    }
  }
}

// ---------------------------------------------------------------------------
// Recurrence: one 512-thread workgroup per batch element.
// Produces X[b*U+u] = [h_u (512) | c_u (512)] and alphas (B,U,T).
// ---------------------------------------------------------------------------
__global__ __launch_bounds__(512)
void recurrence_kernel(const float* __restrict__ enc_out,
                       const int*   __restrict__ enc_mask,
                       const int*   __restrict__ y_in,
                       const float* __restrict__ emb,
                       const float* __restrict__ W_s,
                       const float* __restrict__ vvec,
                       const float* __restrict__ Wih,
                       const float* __restrict__ Whh,
                       const float* __restrict__ bih,
                       const float* __restrict__ bhh,
                       const float* __restrict__ Wh,
                       float* __restrict__ X,
                       float* __restrict__ alphas)
{
  const int b   = blockIdx.x;
  const int tid = threadIdx.x;

  __shared__ float h[DEC_];
  __shared__ float c[ENC_];
  __shared__ float Wsb[ATTN_];
  __shared__ float e[T_];
  __shared__ float alpha[T_];
  __shared__ float embt[EMB_];
  __shared__ float gi[3 * DEC_];
  __shared__ float gh[3 * DEC_];
  __shared__ float red[512];
  __shared__ float sm[2];

  if (tid < DEC_) h[tid] = 0.f;
  __syncthreads();

  for (int u = 0; u < U_; ++u) {
    // ---- Ws = h @ W_s : 2 threads per output column ------------------
    {
      const int a = tid >> 1, half = tid & 1;
      float s = 0.f;
      const int d0 = half * 256;
      for (int d = d0; d < d0 + 256; ++d)
        s += h[d] * W_s[(size_t)d * ATTN_ + a];
      red[tid] = s;
    }
    __syncthreads();
    if ((tid & 1) == 0) Wsb[tid >> 1] = red[tid] + red[tid + 1];
    __syncthreads();

    // ---- e[t] = v . tanh(Wh[b,t,:] + Ws) : 4 threads per t ----------
    {
      const int t = tid >> 2, part = tid & 3;
      const float* whp = Wh + ((size_t)b * T_ + t) * ATTN_;
      float s = 0.f;
      const int a0 = part * 64;
      for (int a = a0; a < a0 + 64; ++a)
        s += vvec[a] * tanhf(whp[a] + Wsb[a]);
      red[tid] = s;
    }
    __syncthreads();
    if (tid < T_) {
      const float s =
          red[tid * 4] + red[tid * 4 + 1] + red[tid * 4 + 2] + red[tid * 4 + 3];
      e[tid] = (enc_mask[b * T_ + tid] == 0) ? -INFINITY : s;
    }
    __syncthreads();

    // ---- softmax over T=128 ------------------------------------------
    if (tid < T_) red[tid] = e[tid];
    __syncthreads();
    for (int s = 64; s > 0; s >>= 1) {
      if (tid < s) red[tid] = fmaxf(red[tid], red[tid + s]);
      __syncthreads();
    }
    if (tid == 0) sm[0] = red[0];
    __syncthreads();
    if (tid < T_) {
      const float ex = expf(e[tid] - sm[0]);
      alpha[tid] = ex;
      red[tid]   = ex;
    }
    __syncthreads();
    for (int s = 64; s > 0; s >>= 1) {
      if (tid < s) red[tid] += red[tid + s];
      __syncthreads();
    }
    if (tid == 0) sm[1] = red[0];
    __syncthreads();
    if (tid < T_) {
      alpha[tid] /= sm[1];
      alphas[((size_t)b * U_ + u) * T_ + tid] = alpha[tid];
    }
    __syncthreads();

    // ---- c = alpha @ enc_out, emb gather -----------------------------
    {
      const int d = tid;                     // 0..511 == ENC
      float s = 0.f;
      for (int t = 0; t < T_; ++t)
        s += alpha[t] * enc_out[((size_t)b * T_ + t) * ENC_ + d];
      c[d] = s;
    }
    if (tid < EMB_) {
      const int y = y_in[b * U_ + u];
      embt[tid] = emb[(size_t)y * EMB_ + tid];
    }
    __syncthreads();

    // ---- GRU pre-activations: 3 gate rows per thread -----------------
#pragma unroll
    for (int q = 0; q < 3; ++q) {
      const int g = tid + q * 512;
      const float* wr = Wih + (size_t)g * (EMB_ + ENC_);
      float s = bih[g];
      for (int j = 0; j < EMB_; ++j) s += embt[j] * wr[j];
      for (int j = 0; j < ENC_; ++j) s += c[j] * wr[EMB_ + j];
      gi[g] = s;
      const float* wr2 = Whh + (size_t)g * DEC_;
      float s2 = bhh[g];
      for (int j = 0; j < DEC_; ++j) s2 += h[j] * wr2[j];
      gh[g] = s2;
    }
    __syncthreads();

    // ---- gates + state update + emit X -------------------------------
    {
      const int d = tid;                     // 0..511 == DEC
      const float r  = 1.f / (1.f + expf(-(gi[d] + gh[d])));
      const float z  = 1.f / (1.f + expf(-(gi[512 + d] + gh[512 + d])));
      const float n  = tanhf(gi[1024 + d] + r * gh[1024 + d]);
      const float hn = (1.f - z) * n + z * h[d];
      float* xp = X + ((size_t)b * U_ + u) * XDIM_;
      xp[d]        = hn;
      xp[DEC_ + d] = c[d];
      h[d] = hn;
    }
    __syncthreads();
  }
}

// ---------------------------------------------------------------------------
// Launcher
// ---------------------------------------------------------------------------
extern "C" void kernel_launch(void* const* d_in, const int* in_sizes, int n_in,
                              void* d_out, int out_size, void* d_ws, size_t ws_size,
                              hipStream_t stream) {
  (void)in_sizes; (void)n_in; (void)out_size; (void)ws_size;

  const float* enc_out = (const float*)d_in[0];
  const int*   enc_mask= (const int*)  d_in[1];
  const int*   y_in    = (const int*)  d_in[2];
  const float* emb     = (const float*)d_in[3];
  const float* W_h     = (const float*)d_in[4];
  const float* W_s     = (const float*)d_in[5];
  const float* v       = (const float*)d_in[6];
  const float* Wih     = (const float*)d_in[7];
  const float* Whh     = (const float*)d_in[8];
  const float* bih     = (const float*)d_in[9];
  const float* bhh     = (const float*)d_in[10];
  const float* out_W   = (const float*)d_in[11];
  const float* out_b   = (const float*)d_in[12];

  // scratch layout
  float* Wh = (float*)d_ws;                               // (B*T, ATTN)  4 MB
  float* X  = Wh + (size_t)B_ * T_ * ATTN_;               // (B*U, 1024)  8 MB

  // output layout: logits (B,U,VOCAB) then alphas (B,U,T)
  float* logits = (float*)d_out;
  float* alphas = logits + (size_t)B_ * U_ * VOCAB_;

  // 1) Wh = enc_out @ W_h      (4096 x 512) @ (512 x 256)
  {
    dim3 grid(ATTN_ / BN, (B_ * T_) / BM);   // (1, 64)
    wmma_gemm_f32<false><<<grid, 256, 0, stream>>>(
        enc_out, W_h, nullptr, Wh, B_ * T_, ATTN_, ENC_);
  }

  // 2) recurrence (attention + GRU), emits X and alphas
  recurrence_kernel<<<B_, 512, 0, stream>>>(
      enc_out, enc_mask, y_in, emb, W_s, v, Wih, Whh, bih, bhh,
      Wh, X, alphas);

  // 3) logits = X @ out_W^T + out_b   (2048 x 1024) @ (1024 x 32000)
  {
    dim3 grid(VOCAB_ / BN, (B_ * U_) / BM);  // (125, 32)
    wmma_gemm_f32<true><<<grid, 256, 0, stream>>>(
        X, out_W, out_b, logits, B_ * U_, VOCAB_, XDIM_);
  }
}